// DirGCNConv_27152783245347
// MI455X (gfx1250) — compile-verified
//
#include <hip/hip_runtime.h>
#include <hip/hip_bf16.h>

#define ALPHA 0.5f
#define D 128              // D_IN == D_OUT == 128

typedef __attribute__((ext_vector_type(2))) float v2f;
typedef __attribute__((ext_vector_type(8))) float v8f;

// ---------------------------------------------------------------------------
// Zero the workspace region (degrees + both aggregation buffers).
// ---------------------------------------------------------------------------
__global__ void dirgcn_zero_kernel(float* p, long long n) {
    long long i = (long long)blockIdx.x * blockDim.x + threadIdx.x;
    long long stride = (long long)gridDim.x * blockDim.x;
    for (; i < n; i += stride) p[i] = 0.0f;
}

// ---------------------------------------------------------------------------
// Degree counting: deg_out[row]++, deg_in[col]++ (as floats).
// ---------------------------------------------------------------------------
__global__ void dirgcn_degree_kernel(const int* __restrict__ ei, int E,
                                     float* __restrict__ deg_out,
                                     float* __restrict__ deg_in) {
    int e = blockIdx.x * blockDim.x + threadIdx.x;
    if (e >= E) return;
    int row = ei[e];
    int col = ei[E + e];
    atomicAdd(&deg_out[row], 1.0f);
    atomicAdd(&deg_in[col], 1.0f);
}

// ---------------------------------------------------------------------------
// deg -> deg^{-1/2} in place (0 for isolated nodes).
// ---------------------------------------------------------------------------
__global__ void dirgcn_invsqrt_kernel(float* __restrict__ deg, int n) {
    int i = blockIdx.x * blockDim.x + threadIdx.x;
    if (i >= n) return;
    float d = deg[i];
    deg[i] = (d > 0.0f) ? rsqrtf(d) : 0.0f;
}

// ---------------------------------------------------------------------------
// Edge scatter: one edge per wave32, 4 dims per lane (float4 gathers, f32
// atomics). Working set (x + agg + edges) ~166MB -> resident in MI455X 192MB L2.
// ---------------------------------------------------------------------------
__global__ void dirgcn_scatter_kernel(const float* __restrict__ x,
                                      const int* __restrict__ ei, int E,
                                      const float* __restrict__ dinv_out,
                                      const float* __restrict__ dinv_in,
                                      float* __restrict__ agg_fwd,
                                      float* __restrict__ agg_bwd) {
    const int lane = threadIdx.x & 31;
    const int e = blockIdx.x * (blockDim.x >> 5) + (threadIdx.x >> 5);
    if (e >= E) return;

    const int row = ei[e];          // wave-uniform
    const int col = ei[E + e];      // wave-uniform
    const float w = dinv_out[row] * dinv_in[col];

    const float4 xc = *(const float4*)(x + (size_t)col * D + lane * 4);
    const float4 xr = *(const float4*)(x + (size_t)row * D + lane * 4);

    float* pf = agg_fwd + (size_t)row * D + lane * 4;
    float* pb = agg_bwd + (size_t)col * D + lane * 4;

    atomicAdd(pf + 0, w * xc.x);
    atomicAdd(pf + 1, w * xc.y);
    atomicAdd(pf + 2, w * xc.z);
    atomicAdd(pf + 3, w * xc.w);

    atomicAdd(pb + 0, w * xr.x);
    atomicAdd(pb + 1, w * xr.y);
    atomicAdd(pb + 2, w * xr.z);
    atomicAdd(pb + 3, w * xr.w);
}

// ---------------------------------------------------------------------------
// WMMA GEMM epilogue (full f32 precision via V_WMMA_F32_16X16X4_F32):
//   out = ALPHA*(agg_fwd @ W_src^T + b_src) + (1-ALPHA)*(agg_bwd @ W_dst^T + b_dst)
// One wave32 per 16x16 output tile; 8 waves/block cover the 8 column tiles of
// D_OUT=128; blockIdx.x covers row tiles (N_NODES % 16 == 0 -> EXEC all ones).
// ---------------------------------------------------------------------------
__global__ __launch_bounds__(256)
void dirgcn_wmma_gemm_kernel(const float* __restrict__ agg_fwd,
                             const float* __restrict__ agg_bwd,
                             const float* __restrict__ W_src,
                             const float* __restrict__ W_dst,
                             const float* __restrict__ b_src,
                             const float* __restrict__ b_dst,
                             float* __restrict__ out) {
    const int lane = threadIdx.x & 31;
    const int waveId = threadIdx.x >> 5;     // 0..7 -> column tile
    const int row0 = blockIdx.x * 16;        // row tile base
    const int col0 = waveId * 16;            // col tile base
    const int half = lane >> 4;              // 0: K pair {0,1}, 1: {2,3}
    const int l15 = lane & 15;

    const int arow = row0 + l15;             // A-matrix row for this lane
    const int ocol = col0 + l15;             // B-matrix column / output column

    // A fragments: A[arow][k + 2*half + {0,1}]
    const float* af = agg_fwd + (size_t)arow * D + half * 2;
    const float* ab = agg_bwd + (size_t)arow * D + half * 2;
    // B fragments: B[k][ocol] = W[ocol][k]; regs hold K = k+2*half, k+2*half+1
    const float* ws = W_src + (size_t)ocol * D + half * 2;
    const float* wd = W_dst + (size_t)ocol * D + half * 2;

    v8f ca = {};
    v8f cb = {};
    #pragma unroll
    for (int k = 0; k < D; k += 4) {
        v2f a0 = *(const v2f*)(af + k);
        v2f b0 = *(const v2f*)(ws + k);
        ca = __builtin_amdgcn_wmma_f32_16x16x4_f32(false, a0, false, b0,
                                                   (short)0, ca, false, false);
        v2f a1 = *(const v2f*)(ab + k);
        v2f b1 = *(const v2f*)(wd + k);
        cb = __builtin_amdgcn_wmma_f32_16x16x4_f32(false, a1, false, b1,
                                                   (short)0, cb, false, false);
    }

    const float bias = ALPHA * b_src[ocol] + (1.0f - ALPHA) * b_dst[ocol];

    // D layout: VGPR v -> M = v (+8 for upper lane half), N = lane & 15
    #pragma unroll
    for (int v = 0; v < 8; ++v) {
        int m = row0 + v + half * 8;
        out[(size_t)m * D + col0 + l15] = ALPHA * ca[v] + (1.0f - ALPHA) * cb[v] + bias;
    }
}

// ---------------------------------------------------------------------------
// Host launcher
// ---------------------------------------------------------------------------
extern "C" void kernel_launch(void* const* d_in, const int* in_sizes, int n_in,
                              void* d_out, int out_size, void* d_ws, size_t ws_size,
                              hipStream_t stream) {
    const float* x     = (const float*)d_in[0];
    const int*   ei    = (const int*)d_in[1];
    const float* W_src = (const float*)d_in[2];
    const float* b_src = (const float*)d_in[3];
    const float* W_dst = (const float*)d_in[4];
    const float* b_dst = (const float*)d_in[5];
    float* out = (float*)d_out;

    const int N = in_sizes[0] / D;       // 100000
    const int E = in_sizes[1] / 2;       // 1600000

    // Workspace layout
    float* ws_f     = (float*)d_ws;
    float* deg_out  = ws_f;                                  // N
    float* deg_in   = ws_f + N;                              // N
    float* agg_fwd  = ws_f + 2 * (size_t)N;                  // N*D
    float* agg_bwd  = ws_f + 2 * (size_t)N + (size_t)N * D;  // N*D

    const long long zcount = 2LL * N + 2LL * N * D;

    // 1) zero degrees + aggregation buffers
    dirgcn_zero_kernel<<<4096, 256, 0, stream>>>(ws_f, zcount);

    // 2) degree counts
    dirgcn_degree_kernel<<<(E + 255) / 256, 256, 0, stream>>>(ei, E, deg_out, deg_in);

    // 3) deg^{-1/2}
    dirgcn_invsqrt_kernel<<<(N + 255) / 256, 256, 0, stream>>>(deg_out, N);
    dirgcn_invsqrt_kernel<<<(N + 255) / 256, 256, 0, stream>>>(deg_in, N);

    // 4) edge scatter (one edge per wave32, 8 waves per block)
    const int edgesPerBlock = 256 / 32;
    dirgcn_scatter_kernel<<<(E + edgesPerBlock - 1) / edgesPerBlock, 256, 0, stream>>>(
        x, ei, E, deg_out, deg_in, agg_fwd, agg_bwd);

    // 5) WMMA GEMM + alpha-blend + bias -> out
    dirgcn_wmma_gemm_kernel<<<N / 16, 256, 0, stream>>>(
        agg_fwd, agg_bwd, W_src, W_dst, b_src, b_dst, out);
}